// ProgressiveGATv2Head_85444079386727
// MI455X (gfx1250) — compile-verified
//
#include <hip/hip_runtime.h>
#include <hip/hip_bf16.h>
#include <math.h>

// ---------------------------------------------------------------------------
// ProgressiveGATv2Head for MI455X (gfx1250, wave32).
// GEMMs use V_WMMA_F32_16X16X4_F32 (full fp32 precision; GEMMs are not the
// bottleneck — edge gather/scatter through L2 is). Each wave computes a
// 16x64 output tile: one A fragment feeds 4 WMMAs per K-step.
// Edge phase: one wave per edge, We staged in LDS, shfl_xor reductions,
// float atomics for segment ops, global prefetch on gathered rows.
// ---------------------------------------------------------------------------

typedef __attribute__((ext_vector_type(2))) float v2f;
typedef __attribute__((ext_vector_type(8))) float v8f;

#define NNODES 20000
#define NEDGES 320000

// ---------------- utility ----------------

__global__ void fill_kernel(float* __restrict__ p, float v, long n) {
  long i = (long)blockIdx.x * blockDim.x + threadIdx.x;
  long st = (long)gridDim.x * blockDim.x;
  for (; i < n; i += st) p[i] = v;
}

__device__ inline float atomicMaxFloat(float* addr, float val) {
  // bit-trick max: ints order like floats for >=0; uints reversed for <0.
  if (val >= 0.0f)
    return __int_as_float(atomicMax((int*)addr, __float_as_int(val)));
  else
    return __uint_as_float(atomicMin((unsigned int*)addr, __float_as_uint(val)));
}

__device__ inline float gelu_exact(float x) {
  return 0.5f * x * (1.0f + erff(x * 0.70710678118654752f));
}

// ---------------- WMMA f32 GEMM: C[M,Nc] = A[M,K] @ B[K,Nc] + bias ----------
// One wave per 16x64 output tile (4 WMMA column-tiles share one A fragment).
// A frag (16x4, ISA 7.12.2): lanes 0-15 hold M=lane,K={k,k+1}; lanes 16-31
// hold M=lane-16,K={k+2,k+3}. B frag (4x16): VGPR0 = rows k (low lanes) /
// k+2 (high lanes); VGPR1 = rows k+1 / k+3. C/D: VGPR i = row i (low lanes)
// and row i+8 (high lanes). Requires M%16==0, Nc%64==0, K%4==0.

__global__ void gemm_wmma_f32(const float* __restrict__ A,
                              const float* __restrict__ B,
                              const float* __restrict__ bias,
                              float* __restrict__ C,
                              int M, int K, int Nc) {
  int wave = threadIdx.x >> 5;
  int lane = threadIdx.x & 31;
  int tilesN = Nc >> 6;                       // 64-wide supertiles
  long tileId = (long)blockIdx.x * (blockDim.x >> 5) + wave;
  long numTiles = (long)(M >> 4) * tilesN;
  if (tileId >= numTiles) return;
  int tm = (int)(tileId / tilesN) << 4;
  int tn = (int)(tileId % tilesN) << 6;
  int half = lane >> 4;      // 0: K pair {k,k+1}, 1: {k+2,k+3}
  int l16  = lane & 15;
  long rowA = tm + l16;
  int  colB = tn + l16;
  v8f acc[4];
#pragma unroll
  for (int t = 0; t < 4; ++t) acc[t] = (v8f){};
  for (int k = 0; k < K; k += 4) {
    int ka = k + half * 2;
    v2f a;
    a.x = A[rowA * K + ka];
    a.y = A[rowA * K + ka + 1];
#pragma unroll
    for (int t = 0; t < 4; ++t) {
      int col = colB + t * 16;
      v2f b;
      b.x = B[(long)ka * Nc + col];
      b.y = B[(long)(ka + 1) * Nc + col];
      acc[t] = __builtin_amdgcn_wmma_f32_16x16x4_f32(false, a, false, b,
                                                     (short)0, acc[t],
                                                     false, false);
    }
  }
  int row0 = tm + half * 8;
#pragma unroll
  for (int t = 0; t < 4; ++t) {
    int col = colB + t * 16;
    float bs = bias ? bias[col] : 0.0f;
#pragma unroll
    for (int i = 0; i < 8; ++i) {
      C[(long)(row0 + i) * Nc + col] = acc[t][i] + bs;
    }
  }
}

// ---------------- column mean of ea (E x 16) --------------------------------

__global__ void col_sum16_kernel(const float* __restrict__ ea,
                                 float* __restrict__ sum16, int E) {
  __shared__ float sh[256];
  int col = threadIdx.x & 15;
  int rloc = threadIdx.x >> 4;  // 0..15
  float acc = 0.0f;
  for (long r = (long)blockIdx.x * 16 + rloc; r < E; r += (long)gridDim.x * 16)
    acc += ea[r * 16 + col];
  sh[threadIdx.x] = acc;
  __syncthreads();
  if (threadIdx.x < 16) {
    float t = 0.0f;
#pragma unroll
    for (int i = 0; i < 16; ++i) t += sh[i * 16 + threadIdx.x];
    atomicAdd(&sum16[threadIdx.x], t);
  }
}

// ee_self[c] = (sum16[k]/E) @ We[k][c]
__global__ void ee_self_kernel(const float* __restrict__ sum16,
                               const float* __restrict__ We,
                               float* __restrict__ ee_self, int hc, float invE) {
  int c = blockIdx.x * blockDim.x + threadIdx.x;
  if (c >= hc) return;
  float s = 0.0f;
#pragma unroll
  for (int k = 0; k < 16; ++k) s += sum16[k] * invE * We[k * hc + c];
  ee_self[c] = s;
}

// ---------------- edge pass 1: scores + segment max -------------------------
// One wave per edge (incl. N self-loop edges appended after the E real ones).
// score[e,h] = sum_c att[h,c]*leaky_relu(xl[src,h,c]+xr[dst,h,c]+ee[e,h,c])
// ee computed on the fly from ea row (regs) x We (LDS); self loops use ee_self.

__global__ void edge_score_kernel(const int* __restrict__ ei,
                                  const float* __restrict__ ea,
                                  const float* __restrict__ We,
                                  const float* __restrict__ att,
                                  const float* __restrict__ xl,
                                  const float* __restrict__ xr,
                                  const float* __restrict__ ee_self,
                                  float* __restrict__ score,
                                  float* __restrict__ mx,
                                  int E, int Nn, int H, int C) {
  const int hc = H * C;
  __shared__ float sWe[16 * 256];
  __shared__ float sAtt[256];
  for (int i = threadIdx.x; i < 16 * hc; i += blockDim.x) sWe[i] = We[i];
  for (int i = threadIdx.x; i < hc; i += blockDim.x) sAtt[i] = att[i];
  __syncthreads();

  int wave = threadIdx.x >> 5;
  int lane = threadIdx.x & 31;
  long e = (long)blockIdx.x * (blockDim.x >> 5) + wave;
  long total = (long)E + Nn;
  if (e >= total) return;

  bool self = (e >= E);
  int src, dst;
  if (self) {
    src = dst = (int)(e - E);
  } else {
    src = ei[e];
    dst = ei[(long)E + e];
  }

  const float* xls = xl + (long)src * hc;
  const float* xrd = xr + (long)dst * hc;
  // kick the random gathers early (global_prefetch_b8 pulls the cachelines)
  __builtin_prefetch(&xls[lane * 8], 0, 3);
  __builtin_prefetch(&xrd[lane * 8], 0, 3);

  float eaR[16];
  if (self) {
#pragma unroll
    for (int k = 0; k < 16; ++k) eaR[k] = 0.0f;
  } else {
#pragma unroll
    for (int k = 0; k < 16; ++k) eaR[k] = ea[e * 16 + k];
  }

  for (int h = 0; h < H; ++h) {
    float s = 0.0f;
    for (int c0 = lane; c0 < C; c0 += 32) {
      int c = h * C + c0;
      float eec;
      if (self) {
        eec = ee_self[c];
      } else {
        eec = 0.0f;
#pragma unroll
        for (int k = 0; k < 16; ++k) eec += eaR[k] * sWe[k * hc + c];
      }
      float v = xls[c] + xrd[c] + eec;
      v = (v > 0.0f) ? v : 0.2f * v;   // leaky_relu 0.2
      s += v * sAtt[c];
    }
#pragma unroll
    for (int m = 16; m >= 1; m >>= 1) s += __shfl_xor(s, m, 32);
    if (lane == 0) {
      score[e * H + h] = s;
      atomicMaxFloat(&mx[(long)dst * H + h], s);
    }
  }
}

// ---------------- edge pass 2: exp + segment sum ----------------------------

__global__ void edge_exp_kernel(const int* __restrict__ ei,
                                float* __restrict__ score,
                                const float* __restrict__ mx,
                                float* __restrict__ sm,
                                int E, int Nn, int H) {
  long idx = (long)blockIdx.x * blockDim.x + threadIdx.x;
  long total = ((long)E + Nn) * H;
  if (idx >= total) return;
  long e = idx / H;
  int h = (int)(idx % H);
  int dst = (e >= E) ? (int)(e - E) : ei[(long)E + e];
  float a = expf(score[idx] - mx[(long)dst * H + h]);
  score[idx] = a;
  atomicAdd(&sm[(long)dst * H + h], a);
}

// ---------------- edge pass 3: weighted scatter-add -------------------------

__global__ void edge_accum_kernel(const int* __restrict__ ei,
                                  const float* __restrict__ score,
                                  const float* __restrict__ sm,
                                  const float* __restrict__ xl,
                                  float* __restrict__ outb,
                                  int E, int Nn, int H, int C) {
  const int hc = H * C;
  int wave = threadIdx.x >> 5;
  int lane = threadIdx.x & 31;
  long e = (long)blockIdx.x * (blockDim.x >> 5) + wave;
  long total = (long)E + Nn;
  if (e >= total) return;
  int src, dst;
  if (e >= E) { src = dst = (int)(e - E); }
  else        { src = ei[e]; dst = ei[(long)E + e]; }
  const float* xls = xl + (long)src * hc;
  float* od = outb + (long)dst * hc;
  __builtin_prefetch(&xls[lane * 8], 0, 3);
  for (int h = 0; h < H; ++h) {
    float w = score[e * H + h] / (sm[(long)dst * H + h] + 1e-16f);
    for (int c0 = lane; c0 < C; c0 += 32) {
      int c = h * C + c0;
      atomicAdd(&od[c], w * xls[c]);
    }
  }
}

// ---------------- LayerNorm + GELU (+pre-bias, +residual) -------------------
// One wave per row; out = gelu(ln(in + pre_bias)*g + b) + res

__global__ void ln_gelu_kernel(const float* __restrict__ in,
                               const float* __restrict__ pre_bias,
                               const float* __restrict__ g,
                               const float* __restrict__ b,
                               const float* __restrict__ res,
                               float* __restrict__ out, int M, int D) {
  int wave = threadIdx.x >> 5;
  int lane = threadIdx.x & 31;
  long row = (long)blockIdx.x * (blockDim.x >> 5) + wave;
  if (row >= M) return;
  float s = 0.0f, s2 = 0.0f;
  for (int d = lane; d < D; d += 32) {
    float v = in[row * D + d] + (pre_bias ? pre_bias[d] : 0.0f);
    s += v;
    s2 += v * v;
  }
#pragma unroll
  for (int m = 16; m >= 1; m >>= 1) {
    s  += __shfl_xor(s, m, 32);
    s2 += __shfl_xor(s2, m, 32);
  }
  float mean = s / (float)D;
  float var  = s2 / (float)D - mean * mean;
  float rstd = rsqrtf(var + 1e-5f);
  for (int d = lane; d < D; d += 32) {
    float v = in[row * D + d] + (pre_bias ? pre_bias[d] : 0.0f);
    float y = (v - mean) * rstd * g[d] + b[d];
    y = gelu_exact(y);
    if (res) y += res[row * D + d];
    out[row * D + d] = y;
  }
}

// ---------------- host orchestration ----------------------------------------

static inline long cdivl(long a, long b) { return (a + b - 1) / b; }

static void launch_gemm(const float* A, const float* B, const float* bias,
                        float* C, int M, int K, int Nc, hipStream_t s) {
  long tiles = (long)(M / 16) * (Nc / 64);
  long blocks = cdivl(tiles, 4);  // 4 waves / block of 128
  gemm_wmma_f32<<<dim3((unsigned)blocks), dim3(128), 0, s>>>(A, B, bias, C, M, K, Nc);
}

static void launch_fill(float* p, float v, long n, hipStream_t s) {
  long blocks = cdivl(n, 256);
  if (blocks > 4096) blocks = 4096;
  fill_kernel<<<dim3((unsigned)blocks), dim3(256), 0, s>>>(p, v, n);
}

extern "C" void kernel_launch(void* const* d_in, const int* in_sizes, int n_in,
                              void* d_out, int out_size, void* d_ws, size_t ws_size,
                              hipStream_t stream) {
  (void)in_sizes; (void)n_in; (void)out_size; (void)ws_size;
  const int Nn = NNODES, E = NEDGES;

  const float* x  = (const float*)d_in[0];
  const int*   ei[3] = {(const int*)d_in[1], (const int*)d_in[3], (const int*)d_in[5]};
  const float* ea[3] = {(const float*)d_in[2], (const float*)d_in[4], (const float*)d_in[6]};
  const float* Wp  = (const float*)d_in[7];
  const float* bp  = (const float*)d_in[8];
  const float* gp  = (const float*)d_in[9];
  const float* bep = (const float*)d_in[10];

  // per-layer params base index
  const float *Wl[3], *bl[3], *Wr[3], *br[3], *We[3], *att[3], *bc[3],
              *gg[3], *be[3], *Wsk[3], *bsk[3];
  for (int i = 0; i < 3; ++i) {
    int o = 11 + i * 11;
    Wl[i]  = (const float*)d_in[o + 0];
    bl[i]  = (const float*)d_in[o + 1];
    Wr[i]  = (const float*)d_in[o + 2];
    br[i]  = (const float*)d_in[o + 3];
    We[i]  = (const float*)d_in[o + 4];
    att[i] = (const float*)d_in[o + 5];
    bc[i]  = (const float*)d_in[o + 6];
    gg[i]  = (const float*)d_in[o + 7];
    be[i]  = (const float*)d_in[o + 8];
    Wsk[i] = (const float*)d_in[o + 9];
    bsk[i] = (const float*)d_in[o + 10];
  }

  static const int IC[3] = {64, 256, 256};
  static const int HH[3] = {4, 4, 1};
  static const int CC[3] = {64, 64, 64};
  static const int OC[3] = {256, 256, 64};

  // workspace layout (floats)
  float* ws   = (float*)d_ws;
  float* hbuf = ws;                         // N*256
  float* h2   = hbuf + (size_t)Nn * 256;    // N*256
  float* xl   = h2   + (size_t)Nn * 256;    // N*256
  float* xr   = xl   + (size_t)Nn * 256;    // N*256
  float* res  = xr   + (size_t)Nn * 256;    // N*256
  float* outb = res  + (size_t)Nn * 256;    // N*256
  float* scr  = outb + (size_t)Nn * 256;    // (E+N)*4
  float* mx   = scr  + (size_t)(E + Nn) * 4;// N*4
  float* smb  = mx   + (size_t)Nn * 4;      // N*4
  float* s16  = smb  + (size_t)Nn * 4;      // 16
  float* eeS  = s16  + 16;                  // 256

  // ---- input projection: h = gelu(ln(x @ Wp + bp)) ----
  launch_gemm(x, Wp, bp, h2, Nn, 64, 64, stream);
  ln_gelu_kernel<<<dim3((unsigned)cdivl(Nn, 8)), dim3(256), 0, stream>>>(
      h2, nullptr, gp, bep, nullptr, hbuf, Nn, 64);

  float* hcur = hbuf;
  float* hnxt = h2;

  for (int i = 0; i < 3; ++i) {
    const int ic = IC[i], H = HH[i], C = CC[i], hc = H * C, oc = OC[i];
    const long totE = (long)E + Nn;

    // dense GEMMs (WMMA f32)
    launch_gemm(hcur, Wl[i], bl[i], xl, Nn, ic, hc, stream);
    launch_gemm(hcur, Wr[i], br[i], xr, Nn, ic, hc, stream);
    launch_gemm(hcur, Wsk[i], bsk[i], res, Nn, ic, oc, stream);

    // self-loop edge feature: mean(ea) @ We
    launch_fill(s16, 0.0f, 16, stream);
    col_sum16_kernel<<<dim3(512), dim3(256), 0, stream>>>(ea[i], s16, E);
    ee_self_kernel<<<dim3((unsigned)cdivl(hc, 256)), dim3(256), 0, stream>>>(
        s16, We[i], eeS, hc, 1.0f / (float)E);

    // init segment buffers
    launch_fill(mx, -INFINITY, (long)Nn * H, stream);
    launch_fill(smb, 0.0f, (long)Nn * H, stream);
    launch_fill(outb, 0.0f, (long)Nn * hc, stream);

    // edge passes
    edge_score_kernel<<<dim3((unsigned)cdivl(totE, 8)), dim3(256), 0, stream>>>(
        ei[i], ea[i], We[i], att[i], xl, xr, eeS, scr, mx, E, Nn, H, C);
    edge_exp_kernel<<<dim3((unsigned)cdivl(totE * H, 256)), dim3(256), 0, stream>>>(
        ei[i], scr, mx, smb, E, Nn, H);
    edge_accum_kernel<<<dim3((unsigned)cdivl(totE, 8)), dim3(256), 0, stream>>>(
        ei[i], scr, smb, xl, outb, E, Nn, H, C);

    // finalize: h' = gelu(ln(att_out + bc)) + res    (concat for i<2; H=1 for i=2)
    float* dest = (i == 2) ? (float*)d_out : hnxt;
    ln_gelu_kernel<<<dim3((unsigned)cdivl(Nn, 8)), dim3(256), 0, stream>>>(
        outb, bc[i], gg[i], be[i], res, dest, Nn, oc);

    // swap h buffers
    float* t = hcur; hcur = hnxt; hnxt = t;
  }
}